// GATModel_39178691674902
// MI455X (gfx1250) — compile-verified
//
#include <hip/hip_runtime.h>
#include <math.h>

// Problem constants (from reference): N=8192, T=60, D_FEAT=6, H=64, 4H=256, 2 heads
#define NSEQ   8192
#define TSTEPS 60
#define DFEAT  6
#define HDIM   64
#define GDIM   256
#define NHEAD  2

typedef float v2f __attribute__((ext_vector_type(2)));
typedef float v8f __attribute__((ext_vector_type(8)));

// D = A(16x4 f32) * B(4x16 f32) + C(16x16 f32), full-precision WMMA.
// A layout (ISA 7.12.2): lane l holds row M=l%16; VGPR r holds K = 4*ks + 2*(l/16) + r.
// B layout (mirror):     lane l holds col N=l%16; VGPR r holds K = 4*ks + 2*(l/16) + r.
// C/D layout:            VGPR r, lane l -> (M = r + 8*(l/16), N = l%16).
__device__ __forceinline__ v8f wmma4(v2f a, v2f b, v8f c) {
  return __builtin_amdgcn_wmma_f32_16x16x4_f32(false, a, false, b, (short)0, c, false, false);
}

__device__ __forceinline__ float sigf(float x)     { return 1.0f / (1.0f + __expf(-x)); }
__device__ __forceinline__ float tanhfast(float x) { return 1.0f - 2.0f / (__expf(2.0f * x) + 1.0f); }
__device__ __forceinline__ float lrelu(float x)    { return x > 0.0f ? x : 0.01f * x; }

// Async global->LDS copy of one 4096-byte tile by 128 threads (2x b128 each).
// ISA 10.x / 15.18.3: GLOBAL_LOAD_ASYNC_TO_LDS_B128, VDST = LDS byte address
// VGPR, VADDR = 64-bit global address, INST_OFFSET added to BOTH addresses.
// Tracked by ASYNCcnt; retire with s_wait_asynccnt.
__device__ __forceinline__ void tile_async_load(const float* __restrict__ gsrc,
                                                const float* ldsdst, int tid) {
  const uint32_t lo = (uint32_t)(uintptr_t)ldsdst + (uint32_t)tid * 16u;
  const char* gp = (const char*)gsrc + tid * 16;
  asm volatile("global_load_async_to_lds_b128 %0, %1, off"
               :: "v"(lo), "v"(gp) : "memory");
  asm volatile("global_load_async_to_lds_b128 %0, %1, off offset:2048"
               :: "v"(lo), "v"(gp) : "memory");
}

// ---------------------------------------------------------------------------
// Kernel 1: fused 2-layer LSTM over all 60 steps.
// One block = 16 sequences, 512 threads = 16 waves; wave w owns gate columns
// [16w, 16w+16). Recurrent weights live in VGPRs as WMMA B-fragments for the
// entire time loop; h/c state and the gate tile live in LDS. Zero HBM traffic
// inside the loop.
// ---------------------------------------------------------------------------
__global__ __launch_bounds__(512) void lstm2_kernel(
    const float* __restrict__ x,
    const float* __restrict__ w_ih0, const float* __restrict__ w_hh0,
    const float* __restrict__ b_ih0, const float* __restrict__ b_hh0,
    const float* __restrict__ w_ih1, const float* __restrict__ w_hh1,
    const float* __restrict__ b_ih1, const float* __restrict__ b_hh1,
    float* __restrict__ h_last)
{
  __shared__ float xls[16 * TSTEPS * DFEAT];   // 23040 B : whole x tile for the block
  __shared__ float wih0_ls[GDIM * DFEAT];      //  6144 B
  __shared__ float bias0_ls[GDIM];             // b_ih0 + b_hh0 folded
  __shared__ float bias1_ls[GDIM];             // b_ih1 + b_hh1 folded
  __shared__ float gates_ls[16 * GDIM];        // 16384 B : gate pre-activations
  __shared__ float h0_ls[16 * HDIM], c0_ls[16 * HDIM];
  __shared__ float h1_ls[16 * HDIM], c1_ls[16 * HDIM];

  const int tid  = threadIdx.x;
  const int lane = tid & 31;
  const int wid  = tid >> 5;        // 0..15 : gate column tile
  const int half = lane >> 4;       // 0/1
  const int l16  = lane & 15;
  const int n0   = blockIdx.x * 16;

  for (int e = tid; e < 16 * TSTEPS * DFEAT; e += 512) xls[e] = x[n0 * (TSTEPS * DFEAT) + e];
  for (int e = tid; e < GDIM * DFEAT;        e += 512) wih0_ls[e] = w_ih0[e];
  for (int e = tid; e < GDIM; e += 512) {
    bias0_ls[e] = b_ih0[e] + b_hh0[e];
    bias1_ls[e] = b_ih1[e] + b_hh1[e];
  }
  for (int e = tid; e < 16 * HDIM; e += 512) {
    h0_ls[e] = 0.0f; c0_ls[e] = 0.0f; h1_ls[e] = 0.0f; c1_ls[e] = 0.0f;
  }

  // Per-wave B fragments: B[k][g] = W[g][k] (gates = h @ W^T).
  const int grow = wid * 16 + l16;  // gate/column index this lane supplies
  v2f whh0B[16], wih1B[16], whh1B[16];
  #pragma unroll
  for (int ks = 0; ks < 16; ++ks) {
    const int k = ks * 4 + 2 * half;
    whh0B[ks][0] = w_hh0[grow * HDIM + k];     whh0B[ks][1] = w_hh0[grow * HDIM + k + 1];
    wih1B[ks][0] = w_ih1[grow * HDIM + k];     wih1B[ks][1] = w_ih1[grow * HDIM + k + 1];
    whh1B[ks][0] = w_hh1[grow * HDIM + k];     whh1B[ks][1] = w_hh1[grow * HDIM + k + 1];
  }
  __syncthreads();

  const int col = grow;

  for (int t = 0; t < TSTEPS; ++t) {
    // ---- layer 0 input GEMM (K=6, VALU) : gates = x_t @ W_ih0^T + bias0 ----
    for (int e = tid; e < 16 * GDIM; e += 512) {
      const int n = e >> 8, g = e & 255;
      float v = bias0_ls[g];
      #pragma unroll
      for (int d = 0; d < DFEAT; ++d)
        v += xls[n * (TSTEPS * DFEAT) + t * DFEAT + d] * wih0_ls[g * DFEAT + d];
      gates_ls[e] = v;
    }
    __syncthreads();

    // ---- layer 0 recurrent GEMM (WMMA): gates += h0 @ W_hh0^T ----
    {
      v8f acc;
      #pragma unroll
      for (int r = 0; r < 8; ++r) acc[r] = gates_ls[(r + 8 * half) * GDIM + col];
      #pragma unroll
      for (int ks = 0; ks < 16; ++ks) {
        const float2 av = *reinterpret_cast<const float2*>(&h0_ls[l16 * HDIM + ks * 4 + 2 * half]);
        v2f a; a[0] = av.x; a[1] = av.y;
        acc = wmma4(a, whh0B[ks], acc);
      }
      #pragma unroll
      for (int r = 0; r < 8; ++r) gates_ls[(r + 8 * half) * GDIM + col] = acc[r];
    }
    __syncthreads();

    // ---- layer 0 gates (i,f,g,o blocks of 64) ----
    for (int e = tid; e < 16 * HDIM; e += 512) {
      const int n = e >> 6, j = e & 63;
      const float gi = gates_ls[n * GDIM + j];
      const float gf = gates_ls[n * GDIM + 64 + j];
      const float gg = gates_ls[n * GDIM + 128 + j];
      const float go = gates_ls[n * GDIM + 192 + j];
      const float c  = sigf(gf) * c0_ls[e] + sigf(gi) * tanhfast(gg);
      c0_ls[e] = c;
      h0_ls[e] = sigf(go) * tanhfast(c);
    }
    __syncthreads();

    // ---- layer 1: gates = bias1 + h0_t @ W_ih1^T + h1_{t-1} @ W_hh1^T (WMMA) ----
    {
      v8f acc;
      const float bv = bias1_ls[col];
      #pragma unroll
      for (int r = 0; r < 8; ++r) acc[r] = bv;
      #pragma unroll
      for (int ks = 0; ks < 16; ++ks) {
        const float2 av = *reinterpret_cast<const float2*>(&h0_ls[l16 * HDIM + ks * 4 + 2 * half]);
        v2f a; a[0] = av.x; a[1] = av.y;
        acc = wmma4(a, wih1B[ks], acc);
      }
      #pragma unroll
      for (int ks = 0; ks < 16; ++ks) {
        const float2 av = *reinterpret_cast<const float2*>(&h1_ls[l16 * HDIM + ks * 4 + 2 * half]);
        v2f a; a[0] = av.x; a[1] = av.y;
        acc = wmma4(a, whh1B[ks], acc);
      }
      #pragma unroll
      for (int r = 0; r < 8; ++r) gates_ls[(r + 8 * half) * GDIM + col] = acc[r];
    }
    __syncthreads();

    // ---- layer 1 gates ----
    for (int e = tid; e < 16 * HDIM; e += 512) {
      const int n = e >> 6, j = e & 63;
      const float gi = gates_ls[n * GDIM + j];
      const float gf = gates_ls[n * GDIM + 64 + j];
      const float gg = gates_ls[n * GDIM + 128 + j];
      const float go = gates_ls[n * GDIM + 192 + j];
      const float c  = sigf(gf) * c1_ls[e] + sigf(gi) * tanhfast(gg);
      c1_ls[e] = c;
      h1_ls[e] = sigf(go) * tanhfast(c);
    }
    __syncthreads();
  }

  for (int e = tid; e < 16 * HDIM; e += 512)
    h_last[n0 * HDIM + e] = h1_ls[e];
}

// ---------------------------------------------------------------------------
// Kernel 2: xt = h_last @ w_t^T + b_t folded directly into the rank-1
// attention projections: s1[hd][n] = xt·a[hd,:64], s2[hd][n] = xt·a[hd,64:].
// xt itself is never stored.
// ---------------------------------------------------------------------------
__global__ __launch_bounds__(256) void attnprep_kernel(
    const float* __restrict__ h_last,
    const float* __restrict__ w_t, const float* __restrict__ b_t,
    const float* __restrict__ a_heads,
    float* __restrict__ s1, float* __restrict__ s2)
{
  __shared__ float wtls[HDIM * HDIM];
  __shared__ float btls[HDIM];
  __shared__ float als[NHEAD * 2 * HDIM];
  const int tid = threadIdx.x;
  for (int e = tid; e < HDIM * HDIM;      e += 256) wtls[e] = w_t[e];
  for (int e = tid; e < HDIM;             e += 256) btls[e] = b_t[e];
  for (int e = tid; e < NHEAD * 2 * HDIM; e += 256) als[e]  = a_heads[e];
  __syncthreads();

  const int n = blockIdx.x * 256 + tid;
  float h[HDIM];
  #pragma unroll
  for (int k = 0; k < HDIM; ++k) h[k] = h_last[n * HDIM + k];

  float s1a0 = 0.f, s1a1 = 0.f, s2a0 = 0.f, s2a1 = 0.f;
  for (int j = 0; j < HDIM; ++j) {
    float xt = btls[j];
    #pragma unroll
    for (int k = 0; k < HDIM; ++k) xt += h[k] * wtls[j * HDIM + k];
    s1a0 += xt * als[j];              s2a0 += xt * als[HDIM + j];
    s1a1 += xt * als[2 * HDIM + j];   s2a1 += xt * als[3 * HDIM + j];
  }
  s1[n] = s1a0;  s1[NSEQ + n] = s1a1;
  s2[n] = s2a0;  s2[NSEQ + n] = s2a1;
}

// ---------------------------------------------------------------------------
// Kernel 2b: global max of s1 per head (exact softmax stabilizer: leaky_relu
// is monotone, so max_j score[i,j] = leaky(s2_i + max_j s1_j)).
// ---------------------------------------------------------------------------
__global__ __launch_bounds__(512) void s1max_kernel(
    const float* __restrict__ s1, float* __restrict__ s1max)
{
  __shared__ float red[1024];
  const int tid = threadIdx.x;
  float m0 = -3.4e38f, m1 = -3.4e38f;
  for (int i = tid; i < NSEQ; i += 512) {
    m0 = fmaxf(m0, s1[i]);
    m1 = fmaxf(m1, s1[NSEQ + i]);
  }
  red[tid] = m0; red[512 + tid] = m1;
  __syncthreads();
  for (int s = 256; s > 0; s >>= 1) {
    if (tid < s) {
      red[tid]       = fmaxf(red[tid],       red[tid + s]);
      red[512 + tid] = fmaxf(red[512 + tid], red[512 + tid + s]);
    }
    __syncthreads();
  }
  if (tid == 0) { s1max[0] = red[0]; s1max[1] = red[512]; }
}

// ---------------------------------------------------------------------------
// Kernel 3: flash-style rank-1 attention (never materializes the 8192x8192
// matrix) + residual + FC + output head, all fused. One block = 16 query
// rows, 128 threads = 4 waves. num += P(16x16) @ H_j(16x64) runs on WMMA.
// The key/value tile is DOUBLE-BUFFERED via GLOBAL_LOAD_ASYNC_TO_LDS_B128:
// the next tile's global->LDS DMA overlaps the current tile's exp/leaky
// probability math and WMMAs, retired with s_wait_asynccnt before the
// consuming barrier. Softmax denominators accumulate via LDS ds_add_f32.
// ---------------------------------------------------------------------------
__global__ __launch_bounds__(128) void attn_kernel(
    const float* __restrict__ h_last,
    const float* __restrict__ s1, const float* __restrict__ s2,
    const float* __restrict__ s1max,
    const float* __restrict__ w_fc, const float* __restrict__ b_fc,
    const float* __restrict__ w_out, const float* __restrict__ b_out,
    float* __restrict__ out)
{
  __shared__ float hjbuf[2][16 * HDIM];      // double-buffered key/value tiles
  __shared__ float pls[NHEAD * 16 * 16];     // probability tiles
  __shared__ float hi[16 * HDIM];            // query-side h_last, then hidden
  __shared__ float numls[NHEAD * 16 * HDIM]; // numerators
  __shared__ float dsum[NHEAD * 16];         // denominators
  __shared__ float s2ls[NHEAD * 16];
  __shared__ float Mls[NHEAD * 16];
  __shared__ float fcls[16 * HDIM];

  const int tid  = threadIdx.x;
  const int lane = tid & 31;
  const int wid  = tid >> 5;        // 0..3
  const int half = lane >> 4;
  const int l16  = lane & 15;
  const int i0   = blockIdx.x * 16;

  const int hd     = wid >> 1;       // head this wave accumulates
  const int ctbase = 2 * (wid & 1);  // first of two 16-wide column tiles

  // Kick off the async DMA of tile 0 before doing anything else.
  tile_async_load(h_last, hjbuf[0], tid);

  for (int e = tid; e < 16 * HDIM; e += 128) hi[e] = h_last[i0 * HDIM + e];
  if (tid < NHEAD * 16) {
    const int h_ = tid >> 4, i = tid & 15;
    const float s2v = s2[h_ * NSEQ + i0 + i];
    s2ls[tid] = s2v;
    Mls[tid]  = lrelu(s2v + s1max[h_]);
    dsum[tid] = 0.0f;
  }
  v8f acc0, acc1;
  #pragma unroll
  for (int r = 0; r < 8; ++r) { acc0[r] = 0.0f; acc1[r] = 0.0f; }
  __syncthreads();

  for (int jt = 0; jt < NSEQ / 16; ++jt) {
    const int j0  = jt * 16;
    const int cur = jt & 1;
    const int nxt = cur ^ 1;
    // Prefetch the next tile (last iteration issues a harmless dummy so the
    // asynccnt level stays uniform; nobody reads that buffer afterwards).
    const int jn = (jt + 1 < NSEQ / 16) ? (jt + 1) : 0;
    tile_async_load(h_last + jn * 16 * HDIM, hjbuf[nxt], tid);

    {
      // 2 heads * 16*16 p-values; 4 consecutive j per thread -> 1 LDS atomic.
      // Independent of hj: hides the async DMA latency.
      const int base = tid * 4;
      const int ph = base >> 8;
      const int pi = (base >> 4) & 15;
      const int pj = base & 15;
      const float s2v = s2ls[ph * 16 + pi];
      const float Mv  = Mls[ph * 16 + pi];
      float lsum = 0.0f;
      #pragma unroll
      for (int q = 0; q < 4; ++q) {
        const float s1v = s1[ph * NSEQ + j0 + pj + q];
        const float p = __expf(lrelu(s2v + s1v) - Mv);
        pls[ph * 256 + pi * 16 + pj + q] = p;
        lsum += p;
      }
      atomicAdd(&dsum[ph * 16 + pi], lsum);
    }

    // Retire tile jt's async loads (2 older + 2 just-issued outstanding; async
    // loads complete in order, so waiting to <=2 guarantees tile jt is in LDS).
    asm volatile("s_wait_asynccnt 0x2" ::: "memory");
    __syncthreads();

    const float* hj = hjbuf[cur];
    #pragma unroll
    for (int ks = 0; ks < 4; ++ks) {
      const int krow = ks * 4 + 2 * half;
      const float2 av = *reinterpret_cast<const float2*>(&pls[hd * 256 + l16 * 16 + krow]);
      v2f a; a[0] = av.x; a[1] = av.y;
      v2f b0, b1;
      b0[0] = hj[(krow    ) * HDIM + ctbase * 16 + l16];
      b0[1] = hj[(krow + 1) * HDIM + ctbase * 16 + l16];
      b1[0] = hj[(krow    ) * HDIM + (ctbase + 1) * 16 + l16];
      b1[1] = hj[(krow + 1) * HDIM + (ctbase + 1) * 16 + l16];
      acc0 = wmma4(a, b0, acc0);
      acc1 = wmma4(a, b1, acc1);
    }
    __syncthreads();
  }

  // spill numerator tiles to LDS (C/D layout)
  #pragma unroll
  for (int r = 0; r < 8; ++r) {
    numls[hd * 16 * HDIM + (r + 8 * half) * HDIM + ctbase * 16 + l16]       = acc0[r];
    numls[hd * 16 * HDIM + (r + 8 * half) * HDIM + (ctbase + 1) * 16 + l16] = acc1[r];
  }
  __syncthreads();

  // hidden = 0.5*(num0/d0 + num1/d1) + h_last   (residual)
  for (int e = tid; e < 16 * HDIM; e += 128) {
    const int i = e >> 6;
    hi[e] = 0.5f * (numls[e] / dsum[i] + numls[16 * HDIM + e] / dsum[16 + i]) + hi[e];
  }
  __syncthreads();

  // fc = leaky(hidden @ w_fc^T + b_fc)
  for (int e = tid; e < 16 * HDIM; e += 128) {
    const int i = e >> 6, j = e & 63;
    float v = b_fc[j];
    #pragma unroll 4
    for (int k2 = 0; k2 < HDIM; ++k2) v += hi[i * HDIM + k2] * w_fc[j * HDIM + k2];
    fcls[e] = lrelu(v);
  }
  __syncthreads();

  // out = fc @ w_out^T + b_out
  if (tid < 16) {
    float v = b_out[0];
    #pragma unroll 4
    for (int j = 0; j < HDIM; ++j) v += fcls[tid * HDIM + j] * w_out[j];
    out[i0 + tid] = v;
  }
}

// ---------------------------------------------------------------------------
extern "C" void kernel_launch(void* const* d_in, const int* in_sizes, int n_in,
                              void* d_out, int out_size, void* d_ws, size_t ws_size,
                              hipStream_t stream) {
  const float* x      = (const float*)d_in[0];
  const float* w_ih0  = (const float*)d_in[1];
  const float* w_hh0  = (const float*)d_in[2];
  const float* b_ih0  = (const float*)d_in[3];
  const float* b_hh0  = (const float*)d_in[4];
  const float* w_ih1  = (const float*)d_in[5];
  const float* w_hh1  = (const float*)d_in[6];
  const float* b_ih1  = (const float*)d_in[7];
  const float* b_hh1  = (const float*)d_in[8];
  const float* w_t    = (const float*)d_in[9];
  const float* b_t    = (const float*)d_in[10];
  const float* a_hds  = (const float*)d_in[11];
  const float* w_fc   = (const float*)d_in[12];
  const float* b_fc   = (const float*)d_in[13];
  const float* w_out  = (const float*)d_in[14];
  const float* b_out  = (const float*)d_in[15];

  float* ws  = (float*)d_ws;
  float* hl  = ws;                          // 8192*64   h_last
  float* s1  = hl + NSEQ * HDIM;            // 2*8192
  float* s2  = s1 + NHEAD * NSEQ;           // 2*8192
  float* s1m = s2 + NHEAD * NSEQ;           // 2

  lstm2_kernel<<<NSEQ / 16, 512, 0, stream>>>(
      x, w_ih0, w_hh0, b_ih0, b_hh0, w_ih1, w_hh1, b_ih1, b_hh1, hl);
  attnprep_kernel<<<NSEQ / 256, 256, 0, stream>>>(hl, w_t, b_t, a_hds, s1, s2);
  s1max_kernel<<<1, 512, 0, stream>>>(s1, s1m);
  attn_kernel<<<NSEQ / 16, 128, 0, stream>>>(
      hl, s1, s2, s1m, w_fc, b_fc, w_out, b_out, (float*)d_out);
}